// TameHighOrderActivationB_85220741087985
// MI455X (gfx1250) — compile-verified
//
#include <hip/hip_runtime.h>
#include <hip/hip_bf16.h>

typedef __attribute__((ext_vector_type(2))) float v2f;
typedef __attribute__((ext_vector_type(8))) float v8f;

// Problem constants (match the reference).
constexpr int kBatch         = 8192;
constexpr int kGroups        = 512;
constexpr int kArity         = 3;
constexpr int kOut           = 16;
constexpr int kVerts         = 27;                 // 3^3
constexpr int kXStride       = kGroups * kArity;   // 1536
constexpr int kOutStride     = kGroups * kOut;     // 8192
constexpr int kGroupsPerWave = 8;
constexpr int kWavesPerBlock = 8;                  // 256 threads = 8 wave32
constexpr int kRowTiles      = kBatch / 16;        // 512
constexpr int kGroupChunks   = kGroups / kGroupsPerWave; // 64

// One wave computes a 16(batch-rows) x 16(out-dim) tile for one group via
// 7x V_WMMA_F32_16X16X4_F32 over the K=27 (padded to 28) basis dimension.
// All 27 basis values are named scalars so nothing ever becomes an indexed
// alloca (no LDS promotion, no scratch) -- K-selection is pure v_cndmask.
__global__ __launch_bounds__(kWavesPerBlock * 32)
void lattice_interp_wmma(const float* __restrict__ X,
                         const float* __restrict__ P,
                         float* __restrict__ out)
{
    const int lane  = threadIdx.x & 31;
    const int wave  = blockIdx.x * kWavesPerBlock + (threadIdx.x >> 5);
    const int rowTile = wave / kGroupChunks;                   // 0..511
    const int gBase   = (wave % kGroupChunks) * kGroupsPerWave;
    const int m   = lane & 15;          // row within tile (A) / column N (B,C/D)
    const bool hi = lane >= 16;         // upper half-wave: K+2/K+3 (A,B), M+8 (C/D)
    const long b  = (long)rowTile * 16 + m;
    const float* xrow = X + b * kXStride;

    for (int gi = 0; gi < kGroupsPerWave; ++gi) {
        const int g = gBase + gi;

        // ---- per-lane inputs for batch row b, group g (streamed: NT loads) ----
        const float x0 = __builtin_nontemporal_load(xrow + g * 3 + 0);
        const float x1 = __builtin_nontemporal_load(xrow + g * 3 + 1);
        const float x2 = __builtin_nontemporal_load(xrow + g * 3 + 2);

        // hat weights over grid {-1,0,+1}: n=relu(-x), z=1-|x|, p=relu(x)
        const float w0n = fmaxf(-x0, 0.f), w0z = 1.f - fabsf(x0), w0p = fmaxf(x0, 0.f);
        const float w1n = fmaxf(-x1, 0.f), w1z = 1.f - fabsf(x1), w1p = fmaxf(x1, 0.f);
        const float w2n = fmaxf(-x2, 0.f), w2z = 1.f - fabsf(x2), w2p = fmaxf(x2, 0.f);

        // pair products q_{d2,d1} = w2[d2]*w1[d1]
        const float qnn = w2n * w1n, qnz = w2n * w1z, qnp = w2n * w1p;
        const float qzn = w2z * w1n, qzz = w2z * w1z, qzp = w2z * w1p;
        const float qpn = w2p * w1n, qpz = w2p * w1z, qpp = w2p * w1p;

        // basis v = d0 + 3*d1 + 9*d2 (dim 0 least significant), as named scalars
        const float b0  = qnn * w0n, b1  = qnn * w0z, b2  = qnn * w0p;
        const float b3  = qnz * w0n, b4  = qnz * w0z, b5  = qnz * w0p;
        const float b6  = qnp * w0n, b7  = qnp * w0z, b8  = qnp * w0p;
        const float b9  = qzn * w0n, b10 = qzn * w0z, b11 = qzn * w0p;
        const float b12 = qzz * w0n, b13 = qzz * w0z, b14 = qzz * w0p;
        const float b15 = qzp * w0n, b16 = qzp * w0z, b17 = qzp * w0p;
        const float b18 = qpn * w0n, b19 = qpn * w0z, b20 = qpn * w0p;
        const float b21 = qpz * w0n, b22 = qpz * w0z, b23 = qpz * w0p;
        const float b24 = qpp * w0n, b25 = qpp * w0z, b26 = qpp * w0p;

        // B-matrix source: params[g][k][n], column n = m for this lane.
        // Half-wave K offset folded into the base pointer; per-chunk offsets
        // are compile-time immediates.
        const float* bp  = P + (long)g * (kVerts * kOut) + m;
        const float* bq  = bp + (hi ? 2 * kOut : 0);            // rows k0, k0+1
        const float* bqe = bp + (hi ? 26 * kOut : 25 * kOut);   // last chunk's 2nd row
                                                                // (hi: K=27 pad -> a==0)
        v8f acc = {};  // C starts at zero

        // A 16x4 layout: lanes 0-15 -> (K, K+1), lanes 16-31 -> (K+2, K+3)
        // B  4x16 layout (symmetric): lane = column N, same K split.
#define DO_CHUNK(K, AL0, AL1, AH0, AH1, B1)                                   \
        do {                                                                  \
            v2f a_, b_;                                                       \
            a_[0] = hi ? (AH0) : (AL0);                                       \
            a_[1] = hi ? (AH1) : (AL1);                                       \
            b_[0] = bq[64 * (K)];                                             \
            b_[1] = (B1);                                                     \
            acc = __builtin_amdgcn_wmma_f32_16x16x4_f32(                      \
                false, a_, false, b_, (short)0, acc, false, false);           \
        } while (0)

        DO_CHUNK(0, b0,  b1,  b2,  b3,   bq[16]);
        DO_CHUNK(1, b4,  b5,  b6,  b7,   bq[80]);
        DO_CHUNK(2, b8,  b9,  b10, b11,  bq[144]);
        DO_CHUNK(3, b12, b13, b14, b15,  bq[208]);
        DO_CHUNK(4, b16, b17, b18, b19,  bq[272]);
        DO_CHUNK(5, b20, b21, b22, b23,  bq[336]);
        DO_CHUNK(6, b24, b25, b26, 0.0f, bqe[0]);
#undef DO_CHUNK

        // C/D 16x16 f32 layout: lane n = lane%16; VGPR r -> M = (hi?8:0)+r.
        // Output is write-once streamed -> non-temporal stores.
        float* orow = out + ((long)rowTile * 16 + (hi ? 8 : 0)) * kOutStride
                          + (long)g * kOut + m;
#pragma unroll
        for (int r = 0; r < 8; ++r)
            __builtin_nontemporal_store(acc[r], orow + (long)r * kOutStride);
    }
}

extern "C" void kernel_launch(void* const* d_in, const int* in_sizes, int n_in,
                              void* d_out, int out_size, void* d_ws, size_t ws_size,
                              hipStream_t stream)
{
    const float* X = (const float*)d_in[0];   // [8192, 1536] f32
    const float* P = (const float*)d_in[1];   // [512, 27, 16] f32
    float* out     = (float*)d_out;           // [8192, 8192] f32

    const int totalWaves = kRowTiles * kGroupChunks;          // 32768
    dim3 grid(totalWaves / kWavesPerBlock);                   // 4096 blocks
    dim3 block(kWavesPerBlock * 32);                          // 256 threads

    hipLaunchKernelGGL(lattice_interp_wmma, grid, block, 0, stream, X, P, out);

    (void)in_sizes; (void)n_in; (void)out_size; (void)d_ws; (void)ws_size;
}